// UCCPredictor_22454089023576
// MI455X (gfx1250) — compile-verified
//
#include <hip/hip_runtime.h>

typedef __attribute__((ext_vector_type(2))) float v2f;
typedef __attribute__((ext_vector_type(8))) float v8f;

#define KDE_NODES 2048
#define KDE_SAMP  512
#define NFEAT     11
#define NB        16
#define BN_EPS    1e-5f

// ---------------------------------------------------------------------------
// Kernel 1: per-(batch,feature) Gaussian KDE, normalized over the 2048 nodes.
// alpha cancels in the normalization, so only exp(beta*d^2) is computed.
// One block per (b,f); 256 threads x 8 nodes each = 2048 nodes.
// ---------------------------------------------------------------------------
__global__ void kde_kernel(const float* __restrict__ x, float* __restrict__ kde) {
    const int bf = blockIdx.x;            // 0..175
    const int b  = bf / NFEAT;
    const int f  = bf % NFEAT;
    const int t  = threadIdx.x;           // 0..255

    __shared__ float xs[KDE_SAMP];
    __shared__ float red[256];

    for (int i = t; i < KDE_SAMP; i += 256)
        xs[i] = x[(size_t)(b * KDE_SAMP + i) * NFEAT + f];
    __syncthreads();

    const float beta = -1.0f / (2.0f * 0.1f * 0.1f);
    const float step = 1.0f / (float)(KDE_NODES - 1);

    float acc[8];
    float sn[8];
#pragma unroll
    for (int j = 0; j < 8; ++j) {
        acc[j] = 0.0f;
        sn[j]  = (float)(t + 256 * j) * step;
    }

    for (int i = 0; i < KDE_SAMP; ++i) {
        const float xv = xs[i];
#pragma unroll
        for (int j = 0; j < 8; ++j) {
            const float d = sn[j] - xv;
            acc[j] += __expf(beta * d * d);
        }
    }

    float tot = 0.0f;
#pragma unroll
    for (int j = 0; j < 8; ++j) tot += acc[j];
    red[t] = tot;
    __syncthreads();
    for (int s = 128; s > 0; s >>= 1) {
        if (t < s) red[t] += red[t + s];
        __syncthreads();
    }
    const float inv = 1.0f / red[0];

    float* out = kde + (size_t)bf * KDE_NODES;
#pragma unroll
    for (int j = 0; j < 8; ++j)
        out[t + 256 * j] = acc[j] * inv;
}

// ---------------------------------------------------------------------------
// Kernel 2/3: conv1d (O=I=11, K=2, stride 2, valid) + eval BatchNorm + ReLU.
// One thread per output element.
// ---------------------------------------------------------------------------
__global__ void conv_bn_relu_kernel(const float* __restrict__ in,
                                    const float* __restrict__ w,
                                    const float* __restrict__ bias,
                                    const float* __restrict__ g,
                                    const float* __restrict__ bb,
                                    const float* __restrict__ m,
                                    const float* __restrict__ v,
                                    float* __restrict__ out,
                                    int Lin, int Lout) {
    const int idx   = blockIdx.x * blockDim.x + threadIdx.x;
    const int total = NB * NFEAT * Lout;
    if (idx >= total) return;

    const int l = idx % Lout;
    const int o = (idx / Lout) % NFEAT;
    const int b = idx / (Lout * NFEAT);

    const float* ib = in + (size_t)b * NFEAT * Lin;
    float acc = bias[o];
#pragma unroll
    for (int c = 0; c < NFEAT; ++c) {
        const float w0 = w[(o * NFEAT + c) * 2 + 0];
        const float w1 = w[(o * NFEAT + c) * 2 + 1];
        acc += ib[c * Lin + 2 * l] * w0 + ib[c * Lin + 2 * l + 1] * w1;
    }
    const float inv = rsqrtf(v[o] + BN_EPS);
    const float y   = (acc - m[o]) * (inv * g[o]) + bb[o];
    out[idx] = fmaxf(y, 0.0f);
}

// ---------------------------------------------------------------------------
// Kernel 4: fully-connected via f32 WMMA.
//   OUT[16, N] = act(H[16, K] @ W[N, K]^T + bias)
// 256 threads = 8 waves; each wave owns one 16-col tile (block covers 128
// cols). K staged through LDS in 64-wide chunks; 16 v_wmma_f32_16x16x4_f32
// per chunk per wave.
//
// Fragment layouts per ISA 7.12.2 (wave32):
//   A 16x4 f32 : lanes 0-15 -> M=lane, VGPR0=K0, VGPR1=K1
//                lanes 16-31 -> M=lane-16, VGPR0=K2, VGPR1=K3
//   B 4x16 f32 : mirrored (N across lanes, K split across lane halves)
//   C/D 16x16  : VGPR r, lanes 0-15 -> M=r, N=lane; lanes 16-31 -> M=r+8
// ---------------------------------------------------------------------------
#define FC_CHUNK 64

__global__ void fc_wmma_kernel(const float* __restrict__ H,    // [16, K]
                               const float* __restrict__ W,    // [N, K]
                               const float* __restrict__ bias, // [N]
                               float* __restrict__ out,        // [16, N]
                               int K, int N, int do_relu) {
    __shared__ float Hs[16][FC_CHUNK + 1];
    __shared__ float Ws[128][FC_CHUNK + 1];

    const int tid     = threadIdx.x;
    const int wave    = tid >> 5;
    const int lane    = tid & 31;
    const int half    = lane >> 4;   // 0: K lo pair, 1: K hi pair
    const int l15     = lane & 15;
    const int colBase = blockIdx.x * 128;
    const int tileCol = colBase + wave * 16;

    v8f acc = {};

    for (int k0 = 0; k0 < K; k0 += FC_CHUNK) {
        for (int e = tid; e < 16 * FC_CHUNK; e += 256) {
            const int r = e / FC_CHUNK, c = e % FC_CHUNK;
            const int k = k0 + c;
            Hs[r][c] = (k < K) ? H[(size_t)r * K + k] : 0.0f;
        }
        for (int e = tid; e < 128 * FC_CHUNK; e += 256) {
            const int r = e / FC_CHUNK, c = e % FC_CHUNK;
            const int n = colBase + r, k = k0 + c;
            Ws[r][c] = (n < N && k < K) ? W[(size_t)n * K + k] : 0.0f;
        }
        __syncthreads();

        if (tileCol < N) {   // wave-uniform: EXEC stays all-1s for WMMA
#pragma unroll
            for (int k = 0; k < FC_CHUNK; k += 4) {
                v2f a, b;
                a.x = Hs[l15][k + half * 2 + 0];
                a.y = Hs[l15][k + half * 2 + 1];
                b.x = Ws[wave * 16 + l15][k + half * 2 + 0];
                b.y = Ws[wave * 16 + l15][k + half * 2 + 1];
                acc = __builtin_amdgcn_wmma_f32_16x16x4_f32(
                    false, a, false, b, (short)0, acc, false, false);
            }
        }
        __syncthreads();
    }

    if (tileCol < N) {
        const int n = tileCol + l15;
        if (n < N) {
            const float bv = bias[n];
#pragma unroll
            for (int r = 0; r < 8; ++r) {
                const int mrow = r + half * 8;
                float y = acc[r] + bv;
                if (do_relu) y = fmaxf(y, 0.0f);
                out[(size_t)mrow * N + n] = y;
            }
        }
    }
}

// ---------------------------------------------------------------------------
extern "C" void kernel_launch(void* const* d_in, const int* in_sizes, int n_in,
                              void* d_out, int out_size, void* d_ws, size_t ws_size,
                              hipStream_t stream) {
    const float* x       = (const float*)d_in[0];
    const float* conv1_w = (const float*)d_in[1];
    const float* conv1_b = (const float*)d_in[2];
    const float* bn1_g   = (const float*)d_in[3];
    const float* bn1_b   = (const float*)d_in[4];
    const float* bn1_m   = (const float*)d_in[5];
    const float* bn1_v   = (const float*)d_in[6];
    const float* conv2_w = (const float*)d_in[7];
    const float* conv2_b = (const float*)d_in[8];
    const float* bn2_g   = (const float*)d_in[9];
    const float* bn2_b   = (const float*)d_in[10];
    const float* bn2_m   = (const float*)d_in[11];
    const float* bn2_v   = (const float*)d_in[12];
    const float* w1      = (const float*)d_in[13];
    const float* b1      = (const float*)d_in[14];
    const float* w2      = (const float*)d_in[15];
    const float* b2      = (const float*)d_in[16];
    const float* w3      = (const float*)d_in[17];
    const float* b3      = (const float*)d_in[18];
    const float* w4      = (const float*)d_in[19];
    const float* b4      = (const float*)d_in[20];

    float* ws  = (float*)d_ws;
    float* kde = ws;                                   // [16*11*2048]
    float* h1  = kde + (size_t)NB * NFEAT * 2048;      // [16*11*1024]
    float* h2  = h1  + (size_t)NB * NFEAT * 1024;      // [16*11*512] == H[16,5632]
    float* f1  = h2  + (size_t)NB * NFEAT * 512;       // [16,512]
    float* f2  = f1  + (size_t)NB * 512;               // [16,128]
    float* f3  = f2  + (size_t)NB * 128;               // [16,32]

    // 1) KDE: 176 blocks (one per (b,f)), in-block normalization.
    kde_kernel<<<NB * NFEAT, 256, 0, stream>>>(x, kde);

    // 2) conv1 + bn1 + relu : [16,11,2048] -> [16,11,1024]
    {
        const int total = NB * NFEAT * 1024;
        conv_bn_relu_kernel<<<(total + 255) / 256, 256, 0, stream>>>(
            kde, conv1_w, conv1_b, bn1_g, bn1_b, bn1_m, bn1_v, h1, 2048, 1024);
    }
    // 3) conv2 + bn2 + relu : [16,11,1024] -> [16,11,512]
    {
        const int total = NB * NFEAT * 512;
        conv_bn_relu_kernel<<<(total + 255) / 256, 256, 0, stream>>>(
            h1, conv2_w, conv2_b, bn2_g, bn2_b, bn2_m, bn2_v, h2, 1024, 512);
    }

    // 4) FC stack via f32 WMMA. reshape(h2) is already [16, 5632] row-major.
    fc_wmma_kernel<<<4, 256, 0, stream>>>(h2, w1, b1, f1, 5632, 512, 1);
    fc_wmma_kernel<<<1, 256, 0, stream>>>(f1, w2, b2, f2,  512, 128, 1);
    fc_wmma_kernel<<<1, 256, 0, stream>>>(f2, w3, b3, f3,  128,  32, 1);
    fc_wmma_kernel<<<1, 256, 0, stream>>>(f3, w4, b4, (float*)d_out, 32, 4, 0);
}